// LSTMRegressor_43069932044465
// MI455X (gfx1250) — compile-verified
//
#include <hip/hip_runtime.h>

// Problem constants (from reference): T=2048, B=1024, H=50
#define T_STEPS 2048
#define BATCH   1024
#define HID     50
#define G4      200        // 4*H gate count
#define GP      208        // gates padded to 13*16
#define KP      52         // K (=H) padded to 13*4
#define NW      13         // one wave per 16-wide gate tile
#define NTHREADS (NW * 32) // 416 threads, wave32

typedef __attribute__((ext_vector_type(2))) float v2f;
typedef __attribute__((ext_vector_type(8))) float v8f;

__device__ __forceinline__ float sigmoid_fast(float v) {
    return __builtin_amdgcn_rcpf(1.0f + __expf(-v));
}
__device__ __forceinline__ float tanh_fast(float v) {
    // tanh(v) = 1 - 2/(exp(2v)+1); exp overflow -> +inf -> rcp -> 0 -> 1 (correct limit)
    return 1.0f - 2.0f * __builtin_amdgcn_rcpf(1.0f + __expf(2.0f * v));
}

__launch_bounds__(NTHREADS, 1)
__global__ void lstm_seq_kernel(const float* __restrict__ x,
                                const float* __restrict__ W_ih,
                                const float* __restrict__ W_hh,
                                const float* __restrict__ b_ih,
                                const float* __restrict__ b_hh,
                                const float* __restrict__ W_lin,
                                const float* __restrict__ b_lin,
                                float* __restrict__ out)
{
    __shared__ float h_lds[16 * KP];   // 16 batch rows x 52 (cols 50,51 stay 0)
    __shared__ float g_lds[16 * GP];   // 16 batch rows x 208 pre-activation gates

    const int tid   = threadIdx.x;
    const int wave  = tid >> 5;
    const int lane  = tid & 31;
    const int n     = lane & 15;   // N-column within tile / M-row for A
    const int khalf = lane >> 4;   // 0: K even-pair base, 1: +2

    // ---- init h = 0 (c lives in registers) ----
    for (int i = tid; i < 16 * KP; i += NTHREADS) h_lds[i] = 0.0f;

    // ---- persistent B fragments: Bmat[k][j] = W_hh[j][k], tile j = wave*16+n ----
    // 4x16 f32 B layout: VGPR0 = rows {k0, k0+2} (lanes 0-15 / 16-31), VGPR1 = {k0+1, k0+3}
    const int j = wave * 16 + n;   // gate column 0..207 (>=200 is zero padding)
    v2f Bfrag[13];
    #pragma unroll
    for (int kk = 0; kk < 13; ++kk) {
        const int k0 = kk * 4 + khalf * 2;
        Bfrag[kk].x = (j < G4 && k0     < HID) ? W_hh[j * HID + k0]     : 0.0f;
        Bfrag[kk].y = (j < G4 && k0 + 1 < HID) ? W_hh[j * HID + k0 + 1] : 0.0f;
    }

    // ---- elementwise-cell thread assignment: elements tid and tid+416 of 800 ----
    const int  b0e  = tid / HID, j0e = tid % HID;        // tid < 416 < 800 always valid
    const int  e1   = tid + NTHREADS;
    const bool has1 = (e1 < 16 * HID);
    const int  b1e  = has1 ? (e1 / HID) : 0, j1e = has1 ? (e1 % HID) : 0;

    float wih0[4], bia0[4], wih1[4], bia1[4];
    #pragma unroll
    for (int q = 0; q < 4; ++q) {
        wih0[q] = W_ih[q * HID + j0e];
        bia0[q] = b_ih[q * HID + j0e] + b_hh[q * HID + j0e];
        wih1[q] = has1 ? W_ih[q * HID + j1e] : 0.0f;
        bia1[q] = has1 ? (b_ih[q * HID + j1e] + b_hh[q * HID + j1e]) : 0.0f;
    }
    float c0 = 0.0f, c1 = 0.0f;

    // x prefetch (batch rows 1008..1023; row 15 == batch 1023 feeds the output)
    const int xbase = BATCH - 16;
    float xcur0 = x[xbase + b0e];
    float xcur1 = has1 ? x[xbase + b1e] : 0.0f;

    // W_lin for the wave-0 output reduction
    float wl0 = 0.0f, wl1 = 0.0f;
    if (wave == 0) {
        wl0 = W_lin[lane];                                  // lane < 32 < 50
        wl1 = (lane + 32 < HID) ? W_lin[lane + 32] : 0.0f;
    }
    const float blin = b_lin[0];

    __syncthreads();

    for (int t = 0; t < T_STEPS; ++t) {
        // ---- out[t-1] = dot(h_{t-1}[row15], W_lin) + b_lin  (overlaps WMMA phase) ----
        if (wave == 0 && t > 0) {
            float p = h_lds[15 * KP + lane] * wl0;
            if (lane + 32 < HID) p += h_lds[15 * KP + lane + 32] * wl1;
            #pragma unroll
            for (int m = 16; m >= 1; m >>= 1) p += __shfl_xor(p, m, 32);
            if (lane == 0) out[t - 1] = p + blin;
        }

        // ---- g = h @ W_hh^T via 13 chained V_WMMA_F32_16X16X4_F32 ----
        // A (16x4 f32): lanes 0-15 = M rows; VGPR0 = K {k0 | k0+2}, VGPR1 = {k0+1 | k0+3}
        v8f acc = {0.f, 0.f, 0.f, 0.f, 0.f, 0.f, 0.f, 0.f};
        #pragma unroll
        for (int kk = 0; kk < 13; ++kk) {
            const int k0 = kk * 4 + khalf * 2;
            v2f A;
            A.x = h_lds[n * KP + k0];
            A.y = h_lds[n * KP + k0 + 1];
            acc = __builtin_amdgcn_wmma_f32_16x16x4_f32(
                false, A, false, Bfrag[kk], (short)0, acc, false, false);
        }
        // D (16x16 f32): VGPR r -> row (r + 8*khalf), col = lane%16 within tile
        #pragma unroll
        for (int r = 0; r < 8; ++r) {
            g_lds[(r + 8 * khalf) * GP + j] = acc[r];
        }
        __syncthreads();

        // ---- elementwise LSTM cell (adds xg = x*W_ih + b_ih + b_hh here) ----
        {
            const int gb = b0e * GP + j0e;
            float gi = g_lds[gb + 0 * HID] + xcur0 * wih0[0] + bia0[0];
            float gf = g_lds[gb + 1 * HID] + xcur0 * wih0[1] + bia0[1];
            float gg = g_lds[gb + 2 * HID] + xcur0 * wih0[2] + bia0[2];
            float go = g_lds[gb + 3 * HID] + xcur0 * wih0[3] + bia0[3];
            c0 = sigmoid_fast(gf) * c0 + sigmoid_fast(gi) * tanh_fast(gg);
            h_lds[b0e * KP + j0e] = sigmoid_fast(go) * tanh_fast(c0);
        }
        if (has1) {
            const int gb = b1e * GP + j1e;
            float gi = g_lds[gb + 0 * HID] + xcur1 * wih1[0] + bia1[0];
            float gf = g_lds[gb + 1 * HID] + xcur1 * wih1[1] + bia1[1];
            float gg = g_lds[gb + 2 * HID] + xcur1 * wih1[2] + bia1[2];
            float go = g_lds[gb + 3 * HID] + xcur1 * wih1[3] + bia1[3];
            c1 = sigmoid_fast(gf) * c1 + sigmoid_fast(gi) * tanh_fast(gg);
            h_lds[b1e * KP + j1e] = sigmoid_fast(go) * tanh_fast(c1);
        }

        // prefetch x for step t+1 (latency hidden behind next WMMA phase)
        if (t + 1 < T_STEPS) {
            xcur0 = x[(t + 1) * BATCH + xbase + b0e];
            if (has1) xcur1 = x[(t + 1) * BATCH + xbase + b1e];
        }
        __syncthreads();
    }

    // ---- final output out[T-1] ----
    if (wave == 0) {
        float p = h_lds[15 * KP + lane] * wl0;
        if (lane + 32 < HID) p += h_lds[15 * KP + lane + 32] * wl1;
        #pragma unroll
        for (int m = 16; m >= 1; m >>= 1) p += __shfl_xor(p, m, 32);
        if (lane == 0) out[T_STEPS - 1] = p + blin;
    }
}

extern "C" void kernel_launch(void* const* d_in, const int* in_sizes, int n_in,
                              void* d_out, int out_size, void* d_ws, size_t ws_size,
                              hipStream_t stream) {
    (void)in_sizes; (void)n_in; (void)d_ws; (void)ws_size; (void)out_size;
    const float* x     = (const float*)d_in[0];
    const float* W_ih  = (const float*)d_in[1];
    const float* W_hh  = (const float*)d_in[2];
    const float* b_ih  = (const float*)d_in[3];
    const float* b_hh  = (const float*)d_in[4];
    const float* W_lin = (const float*)d_in[5];
    const float* b_lin = (const float*)d_in[6];
    float* out = (float*)d_out;

    hipLaunchKernelGGL(lstm_seq_kernel, dim3(1), dim3(NTHREADS), 0, stream,
                       x, W_ih, W_hh, b_ih, b_hh, W_lin, b_lin, out);
}